// GCNencoder_21758304322142
// MI455X (gfx1250) — compile-verified
//
#include <hip/hip_runtime.h>
#include <hip/hip_bf16.h>

typedef float v2f __attribute__((ext_vector_type(2)));
typedef float v8f __attribute__((ext_vector_type(8)));

#define NODES 121
#define NPAD  128
#define H1DIM 32
#define ROWS_PER_WG 128   // 8 waves * 16 batch rows
#define THREADS 256
#define XSTRIDE 132       // dword stride for x/z tile (bank spread for A frags)
#define PSTRIDE 288       // dword stride per K-pair row of adj^T (=2*128 cols + 32 pad)

__global__ __launch_bounds__(THREADS)
void gcn_fused_wmma_kernel(const float* __restrict__ x,
                           const float* __restrict__ adj,
                           const float* __restrict__ W1,
                           const float* __restrict__ W2,
                           float* __restrict__ out) {
    __shared__ float lds_x[NPAD * XSTRIDE];        // per-wave-private 16-row slabs (x, then z)
    __shared__ float lds_b[(NPAD / 2) * PSTRIDE];  // adj^T, K-pair interleaved: [p][n][2]

    const int tid  = threadIdx.x;
    const int wave = tid >> 5;
    const int lane = tid & 31;
    const int mrow = lane & 15;          // M index within 16 (A frag) / N column (B frag)
    const int hi   = lane >> 4;          // 0: lanes 0-15, 1: lanes 16-31
    const int kh   = hi << 1;            // K sub-offset 0 or 2
    const int rbase = wave * 16;         // this wave's private row block in lds_x
    const size_t wg_base = (size_t)blockIdx.x * ROWS_PER_WG;

    // ---- gate constants: Cpos = sum_{W1>0} W1*W2, Cneg = sum_{W1<0} W1*W2 (uniform/SALU) ----
    float cpos = 0.f, cneg = 0.f;
    #pragma unroll
    for (int h = 0; h < H1DIM; ++h) {
        float w1 = W1[h];
        float p  = w1 * W2[h];
        cpos += (w1 > 0.f) ? p : 0.f;
        cneg += (w1 < 0.f) ? p : 0.f;
    }

    // ---- stage x tile (zero-padded) into LDS ----
    for (int i = tid; i < NPAD * NPAD; i += THREADS) {
        int r = i >> 7, c = i & (NPAD - 1);
        float v = 0.f;
        if (c < NODES) v = x[(wg_base + (size_t)r) * NODES + c];
        lds_x[r * XSTRIDE + c] = v;
    }
    // ---- stage adj^T zero-padded, K-pair interleaved: lds_b[(m/2)*PSTRIDE + 2n + (m&1)] = adj[n][m]
    for (int i = tid; i < NPAD * NPAD; i += THREADS) {
        int m = i >> 7, n = i & (NPAD - 1);
        float v = 0.f;
        if (m < NODES && n < NODES) v = adj[n * NODES + m];
        lds_b[(m >> 1) * PSTRIDE + (n << 1) + (m & 1)] = v;
    }
    __syncthreads();

    const int arow = (rbase + mrow) * XSTRIDE;

    // =================== stage 1: Y = X * adj^T ===================
    v8f acc[8] = {};
    #pragma unroll 2
    for (int k = 0; k < NPAD; k += 4) {
        const v2f a = *(const v2f*)&lds_x[arow + k + kh];       // ds_load_b64
        const int p = (k + kh) >> 1;
        #pragma unroll
        for (int nt = 0; nt < 8; ++nt) {
            const int col = nt * 16 + mrow;
            const v2f b = *(const v2f*)&lds_b[p * PSTRIDE + (col << 1)];  // ds_load_b64, pair contiguous
            acc[nt] = __builtin_amdgcn_wmma_f32_16x16x4_f32(
                false, a, false, b, (short)0, acc[nt], false, false);
        }
    }

    // ---- gate: z = y * (y>0 ? Cpos : Cneg); write into this wave's private LDS rows ----
    #pragma unroll
    for (int nt = 0; nt < 8; ++nt) {
        #pragma unroll
        for (int v = 0; v < 8; ++v) {
            float y  = acc[nt][v];
            float zv = y * (y > 0.f ? cpos : cneg);
            int m = v + (hi << 3);
            int n = nt * 16 + mrow;
            lds_x[(rbase + m) * XSTRIDE + n] = zv;   // wave-private rows: no barrier needed
        }
    }

    // =================== stage 2: Out = Z * adj^T ===================
    v8f acc2[8] = {};
    #pragma unroll 2
    for (int k = 0; k < NPAD; k += 4) {
        const v2f a = *(const v2f*)&lds_x[arow + k + kh];
        const int p = (k + kh) >> 1;
        #pragma unroll
        for (int nt = 0; nt < 8; ++nt) {
            const int col = nt * 16 + mrow;
            const v2f b = *(const v2f*)&lds_b[p * PSTRIDE + (col << 1)];
            acc2[nt] = __builtin_amdgcn_wmma_f32_16x16x4_f32(
                false, a, false, b, (short)0, acc2[nt], false, false);
        }
    }

    // ---- store out[b, n] for n < 121 ----
    #pragma unroll
    for (int nt = 0; nt < 8; ++nt) {
        #pragma unroll
        for (int v = 0; v < 8; ++v) {
            int n = nt * 16 + mrow;
            if (n < NODES) {
                int m = v + (hi << 3);
                out[(wg_base + (size_t)(rbase + m)) * NODES + n] = acc2[nt][v];
            }
        }
    }
}

extern "C" void kernel_launch(void* const* d_in, const int* in_sizes, int n_in,
                              void* d_out, int out_size, void* d_ws, size_t ws_size,
                              hipStream_t stream) {
    const float* x   = (const float*)d_in[0];
    const float* adj = (const float*)d_in[1];
    const float* W1  = (const float*)d_in[2];
    const float* W2  = (const float*)d_in[3];
    float* out = (float*)d_out;

    const int B = in_sizes[0] / NODES;                     // 65536
    const int grid = (B + ROWS_PER_WG - 1) / ROWS_PER_WG;  // 512

    gcn_fused_wmma_kernel<<<grid, THREADS, 0, stream>>>(x, adj, W1, W2, out);
}